// MainModel_56504589746874
// MI455X (gfx1250) — compile-verified
//
#include <hip/hip_runtime.h>

// ---------------------------------------------------------------------------
// MI455X plan:
//  - stage 1: per-node neighbor list + degree from edge lists (1024 threads/side)
//  - stage 2: sim = h_s @ h_t^T via V_WMMA_F32_16X16X4_F32 (exact f32, one
//             16x16 tile per wave32, K=64 -> 16 wmma ops/tile, 512 tiles)
//  - stage 3: one wave32 per (i,j): 16x16 Sinkhorn tile in registers,
//             8 elems/lane (lane = half*16+ds, owning 8 dt's); row ops are one
//             shfl_xor(16), column ops are shfl_xor{1,2,4,8}; sim rows for
//             node i cached in LDS with pitch 132 (conflict-free gather).
// Problem is ~1.3MB of traffic (L2-resident) + ~0.8 GFLOP of f32 VALU, so the
// Sinkhorn wave layout is what matters; WMMA handles the (tiny) GEMM exactly.
// ---------------------------------------------------------------------------

#define B_   8
#define N_   128
#define C_   64
#define DEG_ 16
#define NSTEPS_ 5
#define TAU_ 0.1f
#define NEG_ (-1e9f)
#define EPS_ 1e-8f

typedef float v2f __attribute__((ext_vector_type(2)));
typedef float v8f __attribute__((ext_vector_type(8)));

// ---------------- stage 1: dense graph build ----------------
__global__ void build_graph_kernel(const int* __restrict__ edge,  // [2, B*N*DEG]
                                   int* __restrict__ nbr,         // [B*N, DEG]
                                   int* __restrict__ deg)         // [B*N]
{
    int v = blockIdx.x * blockDim.x + threadIdx.x;
    if (v >= B_ * N_) return;
    const int E = B_ * N_ * DEG_;
    const int* dst = edge + E;                 // second row of edge_index
    unsigned w[4] = {0u, 0u, 0u, 0u};
    #pragma unroll
    for (int e = 0; e < DEG_; ++e) {
        int d = dst[v * DEG_ + e] & (N_ - 1);  // local id within graph
        w[d >> 5] |= 1u << (d & 31);
    }
    int cnt = 0;
    #pragma unroll
    for (int wi = 0; wi < 4; ++wi) {
        unsigned x = w[wi];
        while (x) {
            int bpos = __ffs(x) - 1;
            nbr[v * DEG_ + cnt] = wi * 32 + bpos;   // ascending neighbor ids
            ++cnt;
            x &= x - 1;
        }
    }
    deg[v] = cnt;
    for (int p = cnt; p < DEG_; ++p) nbr[v * DEG_ + p] = 0;  // padded (masked out)
}

// ---------------- stage 2: sim = h_s @ h_t^T, f32 WMMA ----------------
__global__ void __launch_bounds__(256)
sim_gemm_kernel(const float* __restrict__ xs,
                const float* __restrict__ xt,
                float* __restrict__ sim)
{
    int wave = threadIdx.x >> 5;
    int lane = threadIdx.x & 31;
    int l    = lane & 15;
    int half = lane >> 4;

    int tile = blockIdx.x * 8 + wave;          // 512 tiles = 8 graphs * 8*8
    int b    = tile >> 6;
    int rem  = tile & 63;
    int i0   = (rem >> 3) << 4;
    int j0   = (rem & 7) << 4;

    // A(16x4 f32) per-lane layout: lanes 0-15 hold (M=l, K=k0,k0+1),
    // lanes 16-31 hold (M=l, K=k0+2,k0+3).  B = h_t^T mirrors A, so both
    // fragments load identically row-major from h_s / h_t.
    const float* arow = xs + (size_t)(b * N_ + i0 + l) * C_ + 2 * half;
    const float* brow = xt + (size_t)(b * N_ + j0 + l) * C_ + 2 * half;

    v8f acc = {};
    #pragma unroll
    for (int k0 = 0; k0 < C_; k0 += 4) {
        v2f a  = *(const v2f*)(arow + k0);
        v2f bb = *(const v2f*)(brow + k0);
        acc = __builtin_amdgcn_wmma_f32_16x16x4_f32(
            /*neg_a=*/false, a, /*neg_b=*/false, bb,
            /*c_mod=*/(short)0, acc, /*reuse_a=*/false, /*reuse_b=*/false);
    }
    // D layout: VGPR r -> (M = r + 8*half, N = l)
    float* outp = sim + ((size_t)b * N_ + i0 + 8 * half) * N_ + j0 + l;
    #pragma unroll
    for (int r = 0; r < 8; ++r) outp[(size_t)r * N_] = acc[r];
}

// ---------------- stage 3: per-pair masked Sinkhorn + update ----------------
__global__ void __launch_bounds__(256)
sinkhorn_kernel(const float* __restrict__ sim,
                const int* __restrict__ nbr_s, const int* __restrict__ deg_s,
                const int* __restrict__ nbr_t, const int* __restrict__ deg_t,
                float* __restrict__ out)
{
    __shared__ float rows[DEG_][N_ + 4];   // pitch 132 -> conflict-free gathers
    __shared__ int   nbrT[N_][DEG_];
    __shared__ int   degT[N_];
    __shared__ int   nbrS[DEG_];
    __shared__ int   degS;

    const int tid = threadIdx.x;
    const int b = blockIdx.x >> 7;
    const int i = blockIdx.x & (N_ - 1);
    const int node_i = b * N_ + i;

    if (tid < DEG_) nbrS[tid] = nbr_s[node_i * DEG_ + tid];
    if (tid == DEG_) degS = deg_s[node_i];
    if (tid >= 32 && tid < 32 + N_) degT[tid - 32] = deg_t[b * N_ + (tid - 32)];
    for (int t = tid; t < N_ * DEG_; t += 256)
        nbrT[t >> 4][t & 15] = nbr_t[(b * N_ + (t >> 4)) * DEG_ + (t & 15)];
    __syncthreads();
    for (int t = tid; t < DEG_ * N_; t += 256) {
        int r = t >> 7, c = t & (N_ - 1);
        rows[r][c] = sim[((size_t)b * N_ + nbrS[r]) * N_ + c];
    }
    __syncthreads();

    const int wave = tid >> 5;
    const int lane = tid & 31;
    const int l    = lane & 15;        // ds row owned by this lane
    const int half = lane >> 4;        // dt half: 0 -> dt 0..7, 1 -> dt 8..15
    const int ds_ok = (l < degS) ? 1 : 0;

    for (int pass = 0; pass < 16; ++pass) {
        const int j   = pass * 8 + wave;
        const int dtj = degT[j];

        float shat[8], S[8], e[8];
        unsigned mbits = 0;
        #pragma unroll
        for (int k = 0; k < 8; ++k) {
            int dt  = half * 8 + k;
            int col = nbrT[j][dt];
            int mk  = ds_ok & ((dt < dtj) ? 1 : 0);
            float v = mk ? rows[l][col] : 0.0f;
            shat[k] = v;
            mbits  |= ((unsigned)mk) << k;
        }

        // masked softmax over dt (16 entries per ds row)
        float mx = NEG_;
        #pragma unroll
        for (int k = 0; k < 8; ++k) {
            float x = ((mbits >> k) & 1u) ? shat[k] : NEG_;
            mx = fmaxf(mx, x);
        }
        mx = fmaxf(mx, __shfl_xor(mx, 16, 32));
        float sum = 0.0f;
        #pragma unroll
        for (int k = 0; k < 8; ++k) {
            float x = ((mbits >> k) & 1u) ? shat[k] : NEG_;
            e[k] = __expf(x - mx);
            sum += e[k];
        }
        sum += __shfl_xor(sum, 16, 32);
        #pragma unroll
        for (int k = 0; k < 8; ++k)
            S[k] = ((mbits >> k) & 1u) ? (e[k] / sum) * (1.0f / TAU_) : 0.0f;

        // 5 Sinkhorn steps: column-normalize (over ds) then row-normalize (over dt)
        for (int it = 0; it < NSTEPS_; ++it) {
            #pragma unroll
            for (int k = 0; k < 8; ++k) {
                float cs = S[k];
                cs += __shfl_xor(cs, 1, 32);
                cs += __shfl_xor(cs, 2, 32);
                cs += __shfl_xor(cs, 4, 32);
                cs += __shfl_xor(cs, 8, 32);   // sum over 16 ds lanes in this half
                S[k] = S[k] / (cs + EPS_);
            }
            float rs = 0.0f;
            #pragma unroll
            for (int k = 0; k < 8; ++k) rs += S[k];
            rs += __shfl_xor(rs, 16, 32);      // partner holds the other 8 dt's
            #pragma unroll
            for (int k = 0; k < 8; ++k) S[k] = S[k] / (rs + EPS_);
        }

        // argmax over dt (first-occurrence tie-break), gather from masked S_hat
        float bv = S[0]; int bk = half * 8; float bs = shat[0];
        #pragma unroll
        for (int k = 1; k < 8; ++k)
            if (S[k] > bv) { bv = S[k]; bk = half * 8 + k; bs = shat[k]; }
        float pv = __shfl_xor(bv, 16, 32);
        int   pk = __shfl_xor(bk, 16, 32);
        float ps = __shfl_xor(bs, 16, 32);
        if (pv > bv || (pv == bv && pk < bk)) { bv = pv; bk = pk; bs = ps; }

        // cost = sum over ds of chosen S_hat entries (16 ds lanes per half)
        float cost = bs;
        cost += __shfl_xor(cost, 1, 32);
        cost += __shfl_xor(cost, 2, 32);
        cost += __shfl_xor(cost, 4, 32);
        cost += __shfl_xor(cost, 8, 32);

        if (lane == 0) {
            float Mv = (float)(degS > dtj ? degS : dtj);
            size_t idx = ((size_t)b * N_ + i) * N_ + j;
            out[idx] = sim[idx] + cost / (1.0f + Mv);
        }
    }
}

// ---------------- launcher ----------------
extern "C" void kernel_launch(void* const* d_in, const int* in_sizes, int n_in,
                              void* d_out, int out_size, void* d_ws, size_t ws_size,
                              hipStream_t stream) {
    const float* xs = (const float*)d_in[0];
    const int*   es = (const int*)d_in[1];
    const float* xt = (const float*)d_in[3];
    const int*   et = (const int*)d_in[4];
    float* out = (float*)d_out;

    char* ws = (char*)d_ws;
    float* sim  = (float*)ws;                                   // 512 KB
    int* nbr_s  = (int*)(ws + (size_t)B_ * N_ * N_ * sizeof(float));
    int* nbr_t  = nbr_s + B_ * N_ * DEG_;
    int* degs   = nbr_t + B_ * N_ * DEG_;
    int* degt   = degs + B_ * N_;

    build_graph_kernel<<<(B_ * N_ + 255) / 256, 256, 0, stream>>>(es, nbr_s, degs);
    build_graph_kernel<<<(B_ * N_ + 255) / 256, 256, 0, stream>>>(et, nbr_t, degt);
    sim_gemm_kernel<<<64, 256, 0, stream>>>(xs, xt, sim);
    sinkhorn_kernel<<<B_ * N_, 256, 0, stream>>>(sim, nbr_s, degs, nbr_t, degt, out);
}